// VIN_20512763806002
// MI455X (gfx1250) — compile-verified
//
#include <hip/hip_runtime.h>

typedef __attribute__((ext_vector_type(2))) float v2f;
typedef __attribute__((ext_vector_type(8))) float v8f;

#define Bsz   128
#define Hdim  64
#define Wdim  64
#define HCH   150
#define QCH   10
#define NACT  8
#define HALO_W 66                    // 64 + 2 halo
#define IMG   (Hdim*Wdim)            // 4096
#define OFF_V0 (HALO_W*HALO_W)       // dword offset of v buffer 0 in smem
#define OFF_V1 (2*HALO_W*HALO_W)     // dword offset of v buffer 1 in smem

// ---------------------------------------------------------------------------
// Kernel 1: r = conv3x3(conv3x3(obs, h_w) + h_b, r_w)
// One workgroup per (batch, 16x16 tile). 150-ch intermediate staged in LDS.
// ---------------------------------------------------------------------------
__global__ void vin_rconv_kernel(const float* __restrict__ obs,
                                 const float* __restrict__ h_w,
                                 const float* __restrict__ h_b,
                                 const float* __restrict__ r_w,
                                 float* __restrict__ r_out) {
    __shared__ float obs_s[2 * 20 * 20];   // obs tile with 2-wide halo
    __shared__ float rimg_s[18 * 18];      // one hidden channel, 1-wide halo
    __shared__ float hw_s[HCH * 18];
    __shared__ float rw_s[HCH * 9];
    __shared__ float hb_s[HCH];

    const int tid  = threadIdx.x;
    const int b    = blockIdx.x >> 4;
    const int tile = blockIdx.x & 15;
    const int ty0  = (tile >> 2) * 16;
    const int tx0  = (tile & 3) * 16;

    for (int i = tid; i < HCH * 18; i += 256) hw_s[i] = h_w[i];
    for (int i = tid; i < HCH * 9;  i += 256) rw_s[i] = r_w[i];
    for (int i = tid; i < HCH;      i += 256) hb_s[i] = h_b[i];

    for (int i = tid; i < 800; i += 256) {
        int ic = i / 400, p = i % 400;
        int py = p / 20,  px = p % 20;
        int gy = ty0 - 2 + py, gx = tx0 - 2 + px;
        float v = 0.f;
        if (gy >= 0 && gy < Hdim && gx >= 0 && gx < Wdim)
            v = obs[((b * 2 + ic) * Hdim + gy) * Wdim + gx];
        obs_s[i] = v;
    }
    __syncthreads();

    const int ty = tid >> 4, tx = tid & 15;
    float racc = 0.f;

    for (int c = 0; c < HCH; ++c) {
        for (int i = tid; i < 324; i += 256) {
            int py = i / 18, px = i % 18;
            int gy = ty0 - 1 + py, gx = tx0 - 1 + px;
            float acc = 0.f;
            if (gy >= 0 && gy < Hdim && gx >= 0 && gx < Wdim) {
                acc = hb_s[c];
                #pragma unroll
                for (int ic = 0; ic < 2; ++ic)
                    #pragma unroll
                    for (int ey = 0; ey < 3; ++ey)
                        #pragma unroll
                        for (int ex = 0; ex < 3; ++ex)
                            acc += hw_s[c * 18 + ic * 9 + ey * 3 + ex] *
                                   obs_s[ic * 400 + (py + ey) * 20 + (px + ex)];
            }
            rimg_s[i] = acc;
        }
        __syncthreads();
        #pragma unroll
        for (int dy = 0; dy < 3; ++dy)
            #pragma unroll
            for (int dx = 0; dx < 3; ++dx)
                racc += rw_s[c * 9 + dy * 3 + dx] * rimg_s[(ty + dy) * 18 + (tx + dx)];
        __syncthreads();
    }
    r_out[b * IMG + (ty0 + ty) * Wdim + (tx0 + tx)] = racc;
}

// Swap the two 16-lane halves of the wave (== __shfl_xor(x, 16, 32)) as a
// pure VALU op: v_permlanex16_b32 with identity nibble selectors.
__device__ __forceinline__ float xor16_f32(float x) {
    int xi = __builtin_bit_cast(int, x);
    int r  = __builtin_amdgcn_permlanex16(xi, xi, 0x76543210, 0xfedcba98,
                                          false, false);
    return __builtin_bit_cast(float, r);
}

// ---------------------------------------------------------------------------
// One value-iteration step for this workgroup's image.
// toffs[10]: per-lane dword offsets into smem (v-taps pre-biased to the
// correct v buffer; K-pad taps alias r-taps whose A weight is zero).
// ---------------------------------------------------------------------------
__device__ __forceinline__ void vin_step(float* __restrict__ smem,
                                         const int toffs[10],
                                         const v2f aw[5],
                                         int voutoff, int wave, int half, int n) {
    const int y0 = wave * 8;                 // 8 rows per wave
    for (int ry = 0; ry < 8; ++ry) {
        const int y       = y0 + ry;
        const int rowbase = y * HALO_W + n;
        const int outrow  = voutoff + (y + 1) * HALO_W + n + 1;
        #pragma unroll
        for (int xt = 0; xt < 4; ++xt) {
            const int base = rowbase + xt * 16;

            // gather all 10 B elements first (unconditional ds_load_b32)
            float bvals[10];
            #pragma unroll
            for (int idx = 0; idx < 10; ++idx)
                bvals[idx] = smem[base + toffs[idx]];

            v8f c = {};
            #pragma unroll
            for (int j = 0; j < 5; ++j) {
                v2f bv;
                bv.x = bvals[2 * j];
                bv.y = bvals[2 * j + 1];
                c = __builtin_amdgcn_wmma_f32_16x16x4_f32(
                        false, aw[j], false, bv, (short)0, c, false, false);
            }
            // branchless max over valid channel rows:
            // lanes 0-15 hold rows 0-7, lanes 16-31 hold rows 8-15 (8,9 valid)
            float m8 = fmaxf(fmaxf(fmaxf(c[0], c[1]), fmaxf(c[2], c[3])),
                             fmaxf(fmaxf(c[4], c[5]), fmaxf(c[6], c[7])));
            float m2 = fmaxf(c[0], c[1]);
            float m  = half ? m2 : m8;
            m = fmaxf(m, xor16_f32(m));
            // lanes n and n+16 hold identical value & address: all lanes store
            smem[outrow + xt * 16] = m;
        }
    }
}

// ---------------------------------------------------------------------------
// Kernel 2: value-iteration loop entirely in LDS, WMMA 16x16x4 f32 stencil.
//   v_{t+1}(x) = max_{a<10} sum_{k<18} Wcat[a,k] * {r,v_t}(x + tap_k)
// GEMM per 16-pixel tile: M=16 (10 channels), K=20 (18 taps), N=16 pixels.
// One workgroup (256 thr = 8 waves) per batch image.
// ---------------------------------------------------------------------------
__global__ void vin_iter_kernel(const float* __restrict__ r_glob,
                                const float* __restrict__ w_i2q,
                                const float* __restrict__ w_v2q,
                                const float* __restrict__ fc_w,
                                const int*   __restrict__ s1,
                                const int*   __restrict__ s2,
                                const int*   __restrict__ kptr,
                                float* __restrict__ logits_out,
                                float* __restrict__ v_out) {
    __shared__ float smem[3 * HALO_W * HALO_W + 16];
    float* r_s  = smem;                       // [0, OFF_V0)  r with halo
    float* qatt = smem + 3 * HALO_W * HALO_W;

    const int tid  = threadIdx.x;
    const int b    = blockIdx.x;
    const int lane = tid & 31;
    const int wave = tid >> 5;
    const int half = lane >> 4;   // 0: C rows 0-7 ; 1: C rows 8-15
    const int n    = lane & 15;   // column (pixel) / A row (channel)

    // zero everything (halo of r and both v buffers must be 0)
    for (int i = tid; i < 3 * HALO_W * HALO_W; i += 256) smem[i] = 0.f;
    __syncthreads();
    for (int i = tid; i < IMG; i += 256) {
        int y = i >> 6, x = i & 63;
        r_s[(y + 1) * HALO_W + (x + 1)] = r_glob[b * IMG + i];
    }

    // ---- per-lane constant A fragments + branchless B tap offsets ----
    // A (16x4 f32): lane = M (=n), VGPR0 holds K = kbase+2*half, VGPR1 = +1.
    v2f aw[5];
    int  toff_even[10];   // v-taps biased by OFF_V0 (vin = v0)
    int  toff_odd[10];    // v-taps biased by OFF_V1 (vin = v1)
    #pragma unroll
    for (int j = 0; j < 5; ++j) {
        #pragma unroll
        for (int p = 0; p < 2; ++p) {
            const int k   = 4 * j + 2 * half + p;
            const int idx = 2 * j + p;
            const int ke  = (k >= 18) ? (k - 18) : k;  // pad taps alias r-taps
            const int cc  = (ke >= 9);
            const int rem = ke - 9 * cc;
            const int off = (rem / 3) * HALO_W + (rem % 3);
            toff_even[idx] = off + (cc ? OFF_V0 : 0);
            toff_odd[idx]  = off + (cc ? OFF_V1 : 0);
            float w = 0.f;
            if (k < 18 && n < QCH)
                w = cc ? w_v2q[n * 9 + rem] : w_i2q[n * 9 + rem];
            if (p == 0) aw[j].x = w; else aw[j].y = w;
        }
    }

    const int iters = kptr[0] + 1;   // k + 1 scan steps

    for (int t = 0; t < iters; ++t) {
        __syncthreads();
        if (t & 1)  // uniform scalar branch: vin = v1 -> vout = v0
            vin_step(smem, toff_odd,  aw, OFF_V0, wave, half, n);
        else        // vin = v0 -> vout = v1
            vin_step(smem, toff_even, aw, OFF_V1, wave, half, n);
    }
    __syncthreads();

    const float* vfin = smem + ((iters & 1) ? OFF_V1 : OFF_V0);

    // v output
    for (int i = tid; i < IMG; i += 256) {
        int y = i >> 6, x = i & 63;
        v_out[b * IMG + i] = vfin[(y + 1) * HALO_W + (x + 1)];
    }

    // q_att at (s1[b], s2[b]); q = conv(cat(r, v_fin), w_cat) at one pixel
    if (tid < QCH) {
        const int sy = s1[b], sx = s2[b];
        float acc = 0.f;
        #pragma unroll
        for (int cch = 0; cch < 2; ++cch) {
            const float* src  = cch ? vfin  : r_s;
            const float* wsrc = cch ? w_v2q : w_i2q;
            #pragma unroll
            for (int dy = 0; dy < 3; ++dy)
                #pragma unroll
                for (int dx = 0; dx < 3; ++dx)
                    acc += wsrc[tid * 9 + dy * 3 + dx] *
                           src[(sy + dy) * HALO_W + (sx + dx)];
        }
        qatt[tid] = acc;
    }
    __syncthreads();
    if (tid < NACT) {
        float lg = 0.f;
        #pragma unroll
        for (int a = 0; a < QCH; ++a) lg += fc_w[tid * QCH + a] * qatt[a];
        logits_out[b * NACT + tid] = lg;
    }
}

// ---------------------------------------------------------------------------
extern "C" void kernel_launch(void* const* d_in, const int* in_sizes, int n_in,
                              void* d_out, int out_size, void* d_ws, size_t ws_size,
                              hipStream_t stream) {
    const float* obs   = (const float*)d_in[0];
    const float* h_w   = (const float*)d_in[1];
    const float* h_b   = (const float*)d_in[2];
    const float* r_w   = (const float*)d_in[3];
    const float* w_i2q = (const float*)d_in[4];
    const float* w_v2q = (const float*)d_in[5];
    const float* fc_w  = (const float*)d_in[6];
    const int*   s1    = (const int*)d_in[7];
    const int*   s2    = (const int*)d_in[8];
    const int*   kptr  = (const int*)d_in[9];

    float* out    = (float*)d_out;
    float* logits = out;                           // (128, 8)
    float* v_out  = out + Bsz * NACT;              // (128, 1, 64, 64)
    float* r_out  = v_out + Bsz * IMG;             // (128, 1, 64, 64)

    vin_rconv_kernel<<<Bsz * 16, 256, 0, stream>>>(obs, h_w, h_b, r_w, r_out);
    vin_iter_kernel<<<Bsz, 256, 0, stream>>>(r_out, w_i2q, w_v2q, fc_w,
                                             s1, s2, kptr, logits, v_out);
}